// DoubleExponentialDisk_76484777607177
// MI455X (gfx1250) — compile-verified
//
#include <hip/hip_runtime.h>
#include <cstdint>

// ---------------------------------------------------------------------------
// DoubleExponentialDisk histogram: 16.7M particles -> 64x64 (r,z) mass grid,
// then divide by annulus volume. Memory-bound scatter: ~268 MB read, ~11.5us
// floor at 23.3 TB/s. Strategy: LDS-privatized histogram (ds_add_f32) with
// bank-shifted replicas, NT streaming b128 loads, global_prefetch, per-block
// partials pushed to d_ws via GLOBAL_STORE_ASYNC_FROM_LDS (CDNA5 async path),
// then a two-stage tree reduction (no global-atomic hotspot).
// ---------------------------------------------------------------------------

#define R_BINS   64
#define Z_BINS   64
#define NBINS    (R_BINS * Z_BINS)      // 4096
#define DR_F     0.15625f               // 10/64, exact
#define DZ_F     0.0625f                // 4/64, exact
#define INV_DR   6.4f
#define INV_DZ   16.0f
#define ZMIN_F   (-2.0f)
#define PI_F     3.14159265358979f

#define REP      2                      // LDS histogram replicas
#define HPAD     32                     // shift replica by 32 dwords -> different banks
#define HSTRIDE  (NBINS + HPAD)         // 4128 floats per replica
#define HTHREADS 256
#define HBLOCKS  1024                   // partial-histogram producers
#define RSLICES  16                     // stage-A reduction slices
#define RCHUNK   (HBLOCKS / RSLICES)    // 64 partials per slice

typedef float v4f __attribute__((ext_vector_type(4)));
typedef int   v4i __attribute__((vector_size(4 * sizeof(int))));
typedef __attribute__((address_space(1))) v4i  g_v4i;   // global int4
typedef __attribute__((address_space(3))) v4i  l_v4i;   // LDS int4
typedef __attribute__((address_space(3))) void las_void; // LDS void (for offset math)

// ---- CDNA5 async store-from-LDS (builtin confirmed present by round-2 diag) ----
__device__ __forceinline__ void async_store_b128(float* gdst, float* lsrc) {
#if __has_builtin(__builtin_amdgcn_global_store_async_from_lds_b128)
    __builtin_amdgcn_global_store_async_from_lds_b128(
        (g_v4i*)gdst, (l_v4i*)lsrc, 0, 0);
#else
    unsigned long long ga = (unsigned long long)(uintptr_t)gdst;
    unsigned           la = (unsigned)(uintptr_t)(las_void*)lsrc; // LDS byte offset
    asm volatile("global_store_async_from_lds_b128 %0, %1, off"
                 :: "v"(ga), "v"(la) : "memory");
#endif
}

__device__ __forceinline__ void wait_asynccnt0() {
#if __has_builtin(__builtin_amdgcn_s_wait_asynccnt)
    __builtin_amdgcn_s_wait_asynccnt(0);
#else
    asm volatile("s_wait_asynccnt 0x0" ::: "memory");
#endif
}

__device__ __forceinline__ void bin_one(float x, float y, float z, float m,
                                        float* __restrict__ h) {
    float r  = __builtin_sqrtf(x * x + y * y);           // r >= 0 always
    float fi = r * INV_DR;                               // >= 0
    float fj = (z - ZMIN_F) * INV_DZ;                    // may be negative
    if (fi < (float)R_BINS && fj >= 0.0f && fj < (float)Z_BINS) {
        int bin = ((int)fi) * Z_BINS + (int)fj;          // trunc == floor (non-neg)
        atomicAdd(&h[bin], m);                           // -> ds_add_f32 (no-return)
    }
}

// MODE 0: push per-block partials (async LDS->global) to dst[blockIdx*NBINS+..]
// MODE 1: global_atomic_add_f32 directly into dst (fallback, fewer blocks)
template <int MODE>
__global__ __launch_bounds__(HTHREADS)
void hist_kernel(const float* __restrict__ pos, const float* __restrict__ mass,
                 float* __restrict__ dst, int N) {
    __shared__ __align__(16) float lds[REP * HSTRIDE];
    for (int t = threadIdx.x; t < REP * HSTRIDE; t += HTHREADS) lds[t] = 0.0f;
    __syncthreads();

    float* h = &lds[(threadIdx.x & (REP - 1)) * HSTRIDE];

    const int quadCount = N >> 2;
    const v4f* __restrict__ pos4 = (const v4f*)pos;
    const v4f* __restrict__ m4   = (const v4f*)mass;
    const int stride = gridDim.x * HTHREADS;

    for (int q = blockIdx.x * HTHREADS + threadIdx.x; q < quadCount; q += stride) {
        int qn = q + stride;
        if (qn < quadCount) {                            // gfx1250 global_prefetch_b8
            __builtin_prefetch(&pos4[3 * (size_t)qn], 0, 0);
            __builtin_prefetch(&m4[qn], 0, 0);
        }
        // 4 particles per iteration: 3x b128 NT position loads + 1x b128 masses
        v4f a = __builtin_nontemporal_load(&pos4[3 * (size_t)q + 0]);
        v4f b = __builtin_nontemporal_load(&pos4[3 * (size_t)q + 1]);
        v4f c = __builtin_nontemporal_load(&pos4[3 * (size_t)q + 2]);
        v4f m = __builtin_nontemporal_load(&m4[q]);
        bin_one(a.x, a.y, a.z, m.x, h);
        bin_one(a.w, b.x, b.y, m.y, h);
        bin_one(b.z, b.w, c.x, m.z, h);
        bin_one(c.y, c.z, c.w, m.w, h);
    }

    // tail (N not multiple of 4) handled by block 0
    if (blockIdx.x == 0) {
        int base = quadCount << 2;
        int rem  = N - base;
        if ((int)threadIdx.x < rem) {
            int p = base + threadIdx.x;
            bin_one(pos[3 * (size_t)p], pos[3 * (size_t)p + 1],
                    pos[3 * (size_t)p + 2], mass[p], h);
        }
    }
    __syncthreads();

    if (MODE == 0) {
        // fold replica 1 into replica 0, then DMA the 16KB histogram out of LDS
        for (int t = threadIdx.x; t < NBINS; t += HTHREADS)
            lds[t] = lds[t] + lds[HSTRIDE + t];
        __syncthreads();
        float* outp = dst + (size_t)blockIdx.x * NBINS;
        for (int c = threadIdx.x; c < NBINS / 4; c += HTHREADS)
            async_store_b128(outp + 4 * c, &lds[4 * c]);  // LDS -> global, no VGPRs
        wait_asynccnt0();
    } else {
        for (int t = threadIdx.x; t < NBINS; t += HTHREADS) {
            float s = lds[t] + lds[HSTRIDE + t];
            (void)__hip_atomic_fetch_add(&dst[t], s, __ATOMIC_RELAXED,
                                         __HIP_MEMORY_SCOPE_AGENT);
        }
    }
}

// Stage A: sum RCHUNK partial histograms per slice; fully coalesced column sums.
__global__ __launch_bounds__(256)
void reduce_stage(const float* __restrict__ partials, float* __restrict__ slice) {
    int bin = blockIdx.x * 256 + threadIdx.x;            // blockIdx.x in [0, NBINS/256)
    int s   = blockIdx.y;                                // [0, RSLICES)
    const float* p = partials + (size_t)s * RCHUNK * NBINS + bin;
    float acc = 0.0f;
#pragma unroll 8
    for (int b = 0; b < RCHUNK; ++b) acc += p[(size_t)b * NBINS];
    slice[(size_t)s * NBINS + bin] = acc;
}

// Stage B: sum slices + divide by annulus volume = pi*DR^2*(2i+1)*DZ.
__global__ __launch_bounds__(256)
void finalize_kernel(const float* __restrict__ slice, float* __restrict__ out) {
    int t = blockIdx.x * 256 + threadIdx.x;
    float acc = 0.0f;
#pragma unroll
    for (int s = 0; s < RSLICES; ++s) acc += slice[(size_t)s * NBINS + t];
    int   i   = t >> 6;
    float vol = PI_F * DR_F * DR_F * (float)(2 * i + 1) * DZ_F;
    out[t] = acc / vol;
}

__global__ __launch_bounds__(256)
void zero_kernel(float* __restrict__ out) {
    out[blockIdx.x * 256 + threadIdx.x] = 0.0f;
}

__global__ __launch_bounds__(256)
void scale_kernel(float* __restrict__ out) {
    int t = blockIdx.x * 256 + threadIdx.x;
    int i = t >> 6;
    float vol = PI_F * DR_F * DR_F * (float)(2 * i + 1) * DZ_F;
    out[t] = out[t] / vol;
}

extern "C" void kernel_launch(void* const* d_in, const int* in_sizes, int n_in,
                              void* d_out, int out_size, void* d_ws, size_t ws_size,
                              hipStream_t stream) {
    const float* pos  = (const float*)d_in[0];   // [N,3] float32
    const float* mass = (const float*)d_in[1];   // [N]   float32
    float* out = (float*)d_out;                  // [64,64] float32
    const int N = in_sizes[1];

    const size_t need = ((size_t)HBLOCKS + RSLICES) * NBINS * sizeof(float);
    if (ws_size >= need) {
        float* partials = (float*)d_ws;
        float* slice    = partials + (size_t)HBLOCKS * NBINS;
        hist_kernel<0><<<HBLOCKS, HTHREADS, 0, stream>>>(pos, mass, partials, N);
        dim3 rg(NBINS / 256, RSLICES);
        reduce_stage<<<rg, 256, 0, stream>>>(partials, slice);
        finalize_kernel<<<NBINS / 256, 256, 0, stream>>>(slice, out);
    } else {
        // Fallback: direct global f32 atomics into d_out (fewer producer blocks
        // to bound the atomic hotspot), then scale in place.
        zero_kernel<<<NBINS / 256, 256, 0, stream>>>(out);
        hist_kernel<1><<<256, HTHREADS, 0, stream>>>(pos, mass, out, N);
        scale_kernel<<<NBINS / 256, 256, 0, stream>>>(out);
    }
}